// OnnxScatterND_11338713661619
// MI455X (gfx1250) — compile-verified
//
#include <hip/hip_runtime.h>
#include <hip/hip_bf16.h>
#include <stdint.h>

// ONNX ScatterND, shapes fixed by the reference:
//   data:    [512, 1024, 256] f32   (512 MiB)
//   indices: [131072, 2] i64        (unique (d0,d1) pairs)
//   updates: [131072, 256] f32      (128 MiB)
//   out:     [512, 1024, 256] f32
//
// Zero-FLOP data movement -> HBM-bandwidth bound on MI455X (23.3 TB/s).
// Row map in d_ws (2 MB), single output pass: each source byte read once,
// each output byte written once; bulk traffic non-temporal (gfx1250 th:NT).
// Total ~1.03 GB vs 1.026 GB floor -> ~44 us roofline.

#define SD0 512
#define SD1 1024
#define SD2 256
#define NUPD 131072
#define NROWS (SD0 * SD1)   // 524288 rows of 1 KiB

typedef float  f32x4 __attribute__((ext_vector_type(4)));
typedef int    i32x4 __attribute__((ext_vector_type(4)));

// ---------------------------------------------------------------------------
// Fallback bulk copy — placed FIRST so the disasm snippet shows its whole
// body and confirms th:TH_LOAD_NT / th:TH_STORE_NT on the b128 pair.
// ---------------------------------------------------------------------------
__global__ void copy_all_kernel(f32x4* __restrict__ o,
                                const f32x4* __restrict__ in) {
    const long long i = (long long)blockIdx.x * blockDim.x + threadIdx.x;
    __builtin_nontemporal_store(__builtin_nontemporal_load(in + i), o + i);
}

// ---------------------------------------------------------------------------
// Phase 2 (hot kernel): one wave32 per 1 KiB row. Source = updates row if
// mapped, else data row. Each lane moves 2x float4 (32 * 2 * 16B = 1024B).
// Plain ternary select: divergence analysis sees r as wave-uniform and the
// round-1 codegen (cndmask select, saddr-form b128) was tighter than the
// readfirstlane/branch variant tried in round 2.
// ---------------------------------------------------------------------------
__global__ void scatter_pass_kernel(float* __restrict__ out,
                                    const float* __restrict__ data,
                                    const float* __restrict__ upd,
                                    const int* __restrict__ map) {
    const int wave = threadIdx.x >> 5;          // 8 waves per block
    const int lane = threadIdx.x & 31;
    const long long r = (long long)blockIdx.x * 8 + wave;   // row id

    const int m = map[r];                       // L2-resident, RT policy
    const float* src = (m >= 0) ? (upd + (long long)m * SD2)
                                : (data + r * SD2);

    const f32x4* s = (const f32x4*)src;
    f32x4*       d = (f32x4*)(out + r * SD2);

    const f32x4 a = __builtin_nontemporal_load(s + lane);
    const f32x4 b = __builtin_nontemporal_load(s + lane + 32);
    __builtin_nontemporal_store(a, d + lane);
    __builtin_nontemporal_store(b, d + lane + 32);
}

// ---------------------------------------------------------------------------
// Phase 1a: map[r] = -1 for all rows (b128 stores; map is re-read next
// kernel and fits easily in the 192 MB L2 -> default RT policy).
// ---------------------------------------------------------------------------
__global__ void init_map_kernel(i32x4* __restrict__ map4) {
    const int i = blockIdx.x * blockDim.x + threadIdx.x;   // exact grid
    i32x4 v; v.x = -1; v.y = -1; v.z = -1; v.w = -1;
    map4[i] = v;
}

// ---------------------------------------------------------------------------
// Phase 1b: map[i0*D1 + i1] = update_index (indices unique per reference).
// ---------------------------------------------------------------------------
__global__ void fill_map_kernel(int* __restrict__ map,
                                const long long* __restrict__ idx) {
    const int i = blockIdx.x * blockDim.x + threadIdx.x;   // exact grid == NUPD
    const long long i0 = idx[2 * i + 0];
    const long long i1 = idx[2 * i + 1];
    map[(int)(i0 * SD1 + i1)] = i;
}

// ---------------------------------------------------------------------------
// Fallback scatter (ws too small path).
// ---------------------------------------------------------------------------
__global__ void scatter_direct_kernel(float* __restrict__ out,
                                      const float* __restrict__ upd,
                                      const long long* __restrict__ idx) {
    const int wave = threadIdx.x >> 5;
    const int lane = threadIdx.x & 31;
    const long long i = (long long)blockIdx.x * 8 + wave;   // update id

    const long long i0 = idx[2 * i + 0];
    const long long i1 = idx[2 * i + 1];
    const long long r  = i0 * SD1 + i1;

    const f32x4* s = (const f32x4*)(upd + i * SD2);
    f32x4*       d = (f32x4*)(out + r * SD2);

    __builtin_nontemporal_store(__builtin_nontemporal_load(s + lane),      d + lane);
    __builtin_nontemporal_store(__builtin_nontemporal_load(s + lane + 32), d + lane + 32);
}

// ---------------------------------------------------------------------------
extern "C" void kernel_launch(void* const* d_in, const int* in_sizes, int n_in,
                              void* d_out, int out_size, void* d_ws, size_t ws_size,
                              hipStream_t stream) {
    const float*     data = (const float*)d_in[0];
    const long long* idx  = (const long long*)d_in[1];
    const float*     upd  = (const float*)d_in[2];
    float*           out  = (float*)d_out;

    if (ws_size >= (size_t)NROWS * sizeof(int)) {
        int* map = (int*)d_ws;
        // 524288 ints -> 131072 int4 stores
        init_map_kernel<<<(NROWS / 4) / 256, 256, 0, stream>>>((i32x4*)map);
        fill_map_kernel<<<NUPD / 256, 256, 0, stream>>>(map, idx);
        // one wave per row, 8 rows per 256-thread block
        scatter_pass_kernel<<<NROWS / 8, 256, 0, stream>>>(out, data, upd, map);
    } else {
        // 134217728 floats -> 33554432 float4, one per thread
        copy_all_kernel<<<(NROWS * (SD2 / 4)) / 256, 256, 0, stream>>>(
            (f32x4*)out, (const f32x4*)data);
        scatter_direct_kernel<<<NUPD / 8, 256, 0, stream>>>(out, upd, idx);
    }
}